// REGI_65077344469385
// MI455X (gfx1250) — compile-verified
//
#include <hip/hip_runtime.h>
#include <math.h>

#define N_NODES 100000
#define E_EDGES 800000
#define NEGO_N  512
#define EEGO_N  4096
#define RREL    5
#define LOG2_C  0.6931471805599453f
#define EPS_C   1e-5f

typedef float v2f __attribute__((ext_vector_type(2)));
typedef float v8f __attribute__((ext_vector_type(8)));

// ---------------------------------------------------------------------------
// Pack B[K x 128] into fragment-paired layout:
//   Bt[(k>>1)*256 + n*2 + (k&1)] = B[k*128 + n]
// so a B-fragment (B[k][n], B[k+1][n]) is one contiguous b64 load per lane.
// ---------------------------------------------------------------------------
__global__ void pack_b(const float* __restrict__ B, float* __restrict__ Bt, int K)
{
    int tid = blockIdx.x * blockDim.x + threadIdx.x;
    if (tid >= K * 128) return;
    int k = tid >> 7, n = tid & 127;
    Bt[(k >> 1) * 256 + n * 2 + (k & 1)] = B[tid];
}

// ---------------------------------------------------------------------------
// C[M x 128] = act( A[M x K] @ B[K x 128] + bias ), K % 8 == 0, M % 16 == 0.
// B passed in packed (pack_b) layout. 256 threads = 8 waves; wave w computes
// rows [blk*128 + 16w, +16) x all 128 cols via V_WMMA_F32_16X16X4_F32.
// k-loop unrolled x2 with all fragments loaded up-front so 18 b64 loads are
// clause-batched and one wait covers 16 wmmas.
// ---------------------------------------------------------------------------
__global__ void gemm_n128_wmma(const float* __restrict__ A, const float* __restrict__ Bt,
                               const float* __restrict__ bias, float* __restrict__ C,
                               int M, int K, int act)
{
    const int wave = threadIdx.x >> 5;
    const int lane = threadIdx.x & 31;
    const int half = lane >> 4;      // 0: lanes 0-15, 1: lanes 16-31
    const int l16  = lane & 15;
    const int row0 = blockIdx.x * 128 + wave * 16;
    if (row0 >= M) return;

    v8f acc[8];
#pragma unroll
    for (int t = 0; t < 8; ++t) acc[t] = (v8f)0.0f;

    const float* Arow = A + (size_t)(row0 + l16) * K;   // this lane's M row

    for (int k0 = 0; k0 < K; k0 += 8) {
        // ---- load all fragments for two k-steps (k0, k0+4) ----
        // A frag (16x4): lanes 0-15 -> K0,K1 ; lanes 16-31 -> K2,K3
        v2f a0 = *(const v2f*)(Arow + k0 + 2 * half);
        v2f a1 = *(const v2f*)(Arow + k0 + 4 + 2 * half);
        const float* Bp0 = Bt + ((k0 + 2 * half) >> 1) * 256 + l16 * 2;
        const float* Bp1 = Bt + ((k0 + 4 + 2 * half) >> 1) * 256 + l16 * 2;
        v2f b0[8], b1[8];
#pragma unroll
        for (int t = 0; t < 8; ++t) b0[t] = *(const v2f*)(Bp0 + t * 32);
#pragma unroll
        for (int t = 0; t < 8; ++t) b1[t] = *(const v2f*)(Bp1 + t * 32);
        // ---- 16 back-to-back wmmas ----
#pragma unroll
        for (int t = 0; t < 8; ++t)
            acc[t] = __builtin_amdgcn_wmma_f32_16x16x4_f32(
                false, a0, false, b0[t], (short)0, acc[t], false, false);
#pragma unroll
        for (int t = 0; t < 8; ++t)
            acc[t] = __builtin_amdgcn_wmma_f32_16x16x4_f32(
                false, a1, false, b1[t], (short)0, acc[t], false, false);
    }
    // C/D layout: VGPR v, lane -> C[row0 + v + 8*half][t*16 + l16]
#pragma unroll
    for (int t = 0; t < 8; ++t) {
        int col = t * 16 + l16;
        float bval = bias ? bias[col] : 0.0f;
#pragma unroll
        for (int v = 0; v < 8; ++v) {
            int r = row0 + v + 8 * half;
            float x = acc[t][v] + bval;
            if (act == 1) x = fmaxf(x, 0.0f);
            C[(size_t)r * 128 + col] = x;
        }
    }
}

// ---------------------------------------------------------------------------
// agg[dst[e]] += hW[etype[e]][src[e]]   (one thread per edge x 4 columns)
// ---------------------------------------------------------------------------
__global__ void edge_scatter(const float* __restrict__ hW, const int* __restrict__ src,
                             const int* __restrict__ dst, const int* __restrict__ et,
                             float* __restrict__ agg)
{
    long long tid = (long long)blockIdx.x * blockDim.x + threadIdx.x;
    long long total = (long long)E_EDGES * 32;
    if (tid >= total) return;
    int e = (int)(tid >> 5);
    int c = (int)(tid & 31) * 4;
    const float4 v = *(const float4*)(hW + ((size_t)et[e] * N_NODES + src[e]) * 128 + c);
    float* out = agg + (size_t)dst[e] * 128 + c;
    atomicAdd(out + 0, v.x);
    atomicAdd(out + 1, v.y);
    atomicAdd(out + 2, v.z);
    atomicAdd(out + 3, v.w);
}

// ---------------------------------------------------------------------------
// In-place leaky_relu(0.01) + per-column sum / sumsq accumulation.
// Total thread count is a multiple of 128 so each thread owns one column.
// ---------------------------------------------------------------------------
__global__ void leaky_stats(float* __restrict__ a, float* __restrict__ stats, int M)
{
    long long tid = (long long)blockIdx.x * blockDim.x + threadIdx.x;
    long long nth = (long long)gridDim.x * blockDim.x;    // multiple of 128
    int col = (int)(tid & 127);
    float s = 0.f, s2 = 0.f;
    long long total = (long long)M * 128;
    for (long long idx = tid; idx < total; idx += nth) {
        float z = a[idx];
        z = (z > 0.f) ? z : 0.01f * z;
        a[idx] = z;
        s += z;
        s2 += z * z;
    }
    atomicAdd(&stats[col], s);
    atomicAdd(&stats[128 + col], s2);
}

// out = relu( (a - mean) * rsqrt(var + eps) ), biased variance (ddof=0)
__global__ void bn_relu(const float* __restrict__ a, const float* __restrict__ stats,
                        float* __restrict__ out, int M)
{
    long long tid = (long long)blockIdx.x * blockDim.x + threadIdx.x;
    if (tid >= (long long)M * 128) return;
    int col = (int)(tid & 127);
    float invM = 1.0f / (float)M;
    float mean = stats[col] * invM;
    float var = stats[128 + col] * invM - mean * mean;
    float x = (a[tid] - mean) * rsqrtf(var + EPS_C);
    out[tid] = fmaxf(x, 0.f);
}

// out[i] = srcMat[ perm ? perm[ids[i]] : ids[i] ]   (rows of 128)
__global__ void gather_rows(const float* __restrict__ srcMat, const int* __restrict__ ids,
                            const int* __restrict__ perm, float* __restrict__ out, int n)
{
    int tid = blockIdx.x * blockDim.x + threadIdx.x;
    if (tid >= n * 128) return;
    int i = tid >> 7, c = tid & 127;
    int r = ids[i];
    if (perm) r = perm[r];
    out[tid] = srcMat[(size_t)r * 128 + c];
}

__global__ void build_adj(const int* __restrict__ es, const int* __restrict__ ed,
                          unsigned* __restrict__ adjbits)
{
    int e = blockIdx.x * blockDim.x + threadIdx.x;
    if (e >= EEGO_N) return;
    int s = es[e], d = ed[e];
    atomicOr(&adjbits[s * 16 + (d >> 5)], 1u << (d & 31));
}

// ---------------------------------------------------------------------------
// Sequential 512-step push/pull scan. grid=2 (pos,neg), 128 threads (1 column
// each). m[512][128] lives in 256KB dynamic LDS; root state compressed to a
// 512-entry index chain (root rows are always copies of root0 rows).
// ---------------------------------------------------------------------------
__global__ void disc_scan(const float* __restrict__ fcx, const float* __restrict__ Wfc,
                          const float* __restrict__ bfc, const unsigned* __restrict__ adjbits,
                          float* __restrict__ snap_m, int* __restrict__ snap_ri)
{
    extern __shared__ float m[];                 // [512][128]
    __shared__ int rootidx[NEGO_N];
    const int h = threadIdx.x;                   // column 0..127
    float* sm_out = snap_m + (size_t)blockIdx.x * NEGO_N * 128;
    int*   ri_out = snap_ri + blockIdx.x * NEGO_N;

    for (int j = 0; j < NEGO_N; ++j) m[j * 128 + h] = 0.f;
    for (int j = h; j < NEGO_N; j += 128) rootidx[j] = j;
    __syncthreads();

    for (int i = 0; i < NEGO_N; ++i) {
        // snapshot (apply_edges state) before push
        sm_out[i * 128 + h] = m[i * 128 + h];
        if (h == 0) ri_out[i] = rootidx[i];
        // message
        float msg;
        if (i == 1) {
            msg = fcx[128 + h];                   // depth-1 message fc(x[1])
        } else {
            float acc = bfc[h];
            for (int k = 0; k < 128; ++k)
                acc += m[i * 128 + k] * Wfc[k * 128 + h];  // LDS broadcast reads
            msg = acc;
        }
        int newri = rootidx[i];
        __syncthreads();                          // finish reading m[i] before writes
        // push to out-neighbors
        for (int w = 0; w < 16; ++w) {
            unsigned bits = adjbits[i * 16 + w];
            while (bits) {
                int b = __ffs(bits) - 1;
                bits &= bits - 1;
                int j = w * 32 + b;
                m[j * 128 + h] = fmaxf(fcx[j * 128 + h] + msg, 0.f);
                if (h == 0) rootidx[j] = newri;
            }
        }
        __syncthreads();
    }
}

// hcat[run][e] = [ root0[run][snap_ri[run][src]] | snap_m[run][src] | xego[dst] ]
__global__ void build_hcat(const float* __restrict__ root0, const float* __restrict__ snap_m,
                           const int* __restrict__ snap_ri, const float* __restrict__ xego,
                           const int* __restrict__ es, const int* __restrict__ ed,
                           float* __restrict__ hcat)
{
    int tid = blockIdx.x * blockDim.x + threadIdx.x;
    if (tid >= 2 * EEGO_N * 128) return;
    int c = tid & 127;
    int t = tid >> 7;                 // 0..8191
    int run = t / EEGO_N;             // 0 = pos, 1 = neg
    int e = t - run * EEGO_N;
    int s = es[e], d = ed[e];
    size_t base = (size_t)t * 384;
    const float* r0 = root0 + ((size_t)run * NEGO_N + snap_ri[run * NEGO_N + s]) * 128;
    hcat[base + c]       = r0[c];
    hcat[base + 128 + c] = snap_m[((size_t)run * NEGO_N + s) * 128 + c];
    hcat[base + 256 + c] = xego[(size_t)d * 128 + c];
}

// logit = Hrelu[row]·Wus + bus ; accumulate JSD terms into sums[0]=pos, sums[1]=neg
__global__ void logits_reduce(const float* __restrict__ Hrelu, const float* __restrict__ Wus,
                              const float* __restrict__ bus, float* __restrict__ sums)
{
    int lane = threadIdx.x & 31;
    int wave = threadIdx.x >> 5;
    int row = blockIdx.x * 8 + wave;
    if (row >= 2 * EEGO_N) return;
    const float* hr = Hrelu + (size_t)row * 128;
    float acc = 0.f;
    for (int k = lane; k < 128; k += 32) acc += hr[k] * Wus[k];
    for (int off = 16; off; off >>= 1) acc += __shfl_down(acc, off, 32);
    if (lane == 0) {
        float logit = acc + bus[0];
        // softplus(-logit), numerically stable
        float sp = fmaxf(-logit, 0.f) + log1pf(expf(-fabsf(logit)));
        if (row < EEGO_N) atomicAdd(&sums[0], LOG2_C - sp);
        else              atomicAdd(&sums[1], sp + logit - LOG2_C);
    }
}

__global__ void finalize(const float* __restrict__ sums, float* __restrict__ out)
{
    if (threadIdx.x == 0 && blockIdx.x == 0)
        out[0] = (sums[1] - sums[0]) / (float)EEGO_N;
}

// ---------------------------------------------------------------------------
extern "C" void kernel_launch(void* const* d_in, const int* in_sizes, int n_in,
                              void* d_out, int out_size, void* d_ws, size_t ws_size,
                              hipStream_t stream)
{
    const float* features = (const float*)d_in[0];
    const float* W1      = (const float*)d_in[1];
    const float* Wloop1  = (const float*)d_in[2];
    const float* b1      = (const float*)d_in[3];
    const float* W2      = (const float*)d_in[4];
    const float* Wloop2  = (const float*)d_in[5];
    const float* b2      = (const float*)d_in[6];
    const float* Wfc     = (const float*)d_in[7];
    const float* bfc     = (const float*)d_in[8];
    const float* Wlin    = (const float*)d_in[9];
    const float* blin    = (const float*)d_in[10];
    const float* Wus     = (const float*)d_in[11];
    const float* bus     = (const float*)d_in[12];
    const int* src       = (const int*)d_in[13];
    const int* dst       = (const int*)d_in[14];
    const int* etype     = (const int*)d_in[15];
    const int* perm      = (const int*)d_in[16];
    const int* ego_ids   = (const int*)d_in[17];
    const int* ego_src   = (const int*)d_in[18];
    const int* ego_dst   = (const int*)d_in[19];

    float* ws = (float*)d_ws;
    size_t o = 0;
    float* hW    = ws + o; o += (size_t)RREL * N_NODES * 128;   // 256 MB
    float* agg   = ws + o; o += (size_t)N_NODES * 128;
    float* h1    = ws + o; o += (size_t)N_NODES * 128;
    float* h2    = ws + o; o += (size_t)N_NODES * 128;
    float* stats = ws + o; o += 256;
    float* xego  = ws + o; o += NEGO_N * 128;
    float* root0 = ws + o; o += 2 * NEGO_N * 128;               // [pos|neg]
    float* fcx   = ws + o; o += NEGO_N * 128;
    unsigned* adjbits = (unsigned*)(ws + o); o += NEGO_N * 16;
    float* snapm = ws + o; o += 2 * NEGO_N * 128;
    int*   snapri= (int*)(ws + o); o += 2 * NEGO_N;
    float* hcat  = ws + o; o += (size_t)2 * EEGO_N * 384;
    float* Hrelu = ws + o; o += (size_t)2 * EEGO_N * 128;
    float* sums  = ws + o; o += 2;
    // packed weight matrices (fragment-paired layout)
    float* BtWloop1 = ws + o; o += 16384;
    float* BtW1     = ws + o; o += (size_t)RREL * 16384;
    float* BtWloop2 = ws + o; o += 16384;
    float* BtW2     = ws + o; o += (size_t)RREL * 16384;
    float* BtWfc    = ws + o; o += 16384;
    float* BtWlin   = ws + o; o += 384 * 128;

    hipMemsetAsync(stats, 0, 256 * sizeof(float), stream);
    hipMemsetAsync(adjbits, 0, NEGO_N * 16 * sizeof(unsigned), stream);
    hipMemsetAsync(sums, 0, 2 * sizeof(float), stream);

    dim3 blk(256);
    const int pb128 = (128 * 128 + 255) / 256;
    pack_b<<<pb128, blk, 0, stream>>>(Wloop1, BtWloop1, 128);
    pack_b<<<pb128, blk, 0, stream>>>(Wloop2, BtWloop2, 128);
    pack_b<<<pb128, blk, 0, stream>>>(Wfc, BtWfc, 128);
    pack_b<<<(384 * 128 + 255) / 256, blk, 0, stream>>>(Wlin, BtWlin, 384);
    for (int r = 0; r < RREL; ++r) {
        pack_b<<<pb128, blk, 0, stream>>>(W1 + (size_t)r * 16384, BtW1 + (size_t)r * 16384, 128);
        pack_b<<<pb128, blk, 0, stream>>>(W2 + (size_t)r * 16384, BtW2 + (size_t)r * 16384, 128);
    }

    const int gemmBlocksN = (N_NODES + 127) / 128;
    const int edgeBlocks  = (int)(((long long)E_EDGES * 32 + 255) / 256);
    const int nodeBlocks  = (int)(((long long)N_NODES * 128 + 255) / 256);

    // ---- Layer 1: agg = features@Wloop1 + b1 ; += segment_sum(hW[etype,src]) ----
    gemm_n128_wmma<<<gemmBlocksN, blk, 0, stream>>>(features, BtWloop1, b1, agg, N_NODES, 128, 0);
    for (int r = 0; r < RREL; ++r)
        gemm_n128_wmma<<<gemmBlocksN, blk, 0, stream>>>(
            features, BtW1 + (size_t)r * 16384, nullptr,
            hW + (size_t)r * N_NODES * 128, N_NODES, 128, 0);
    edge_scatter<<<edgeBlocks, blk, 0, stream>>>(hW, src, dst, etype, agg);
    leaky_stats<<<1024, blk, 0, stream>>>(agg, stats, N_NODES);
    bn_relu<<<nodeBlocks, blk, 0, stream>>>(agg, stats, h1, N_NODES);

    // ---- Layer 2 ----
    hipMemsetAsync(stats, 0, 256 * sizeof(float), stream);
    gemm_n128_wmma<<<gemmBlocksN, blk, 0, stream>>>(h1, BtWloop2, b2, agg, N_NODES, 128, 0);
    for (int r = 0; r < RREL; ++r)
        gemm_n128_wmma<<<gemmBlocksN, blk, 0, stream>>>(
            h1, BtW2 + (size_t)r * 16384, nullptr,
            hW + (size_t)r * N_NODES * 128, N_NODES, 128, 0);
    edge_scatter<<<edgeBlocks, blk, 0, stream>>>(hW, src, dst, etype, agg);
    leaky_stats<<<1024, blk, 0, stream>>>(agg, stats, N_NODES);
    bn_relu<<<nodeBlocks, blk, 0, stream>>>(agg, stats, h2, N_NODES);

    // ---- Discriminator ----
    const int egoBlocks = (NEGO_N * 128 + 255) / 256;
    gather_rows<<<egoBlocks, blk, 0, stream>>>(features, ego_ids, nullptr, xego, NEGO_N);
    gather_rows<<<egoBlocks, blk, 0, stream>>>(h2, ego_ids, nullptr, root0, NEGO_N);             // pos
    gather_rows<<<egoBlocks, blk, 0, stream>>>(h2, ego_ids, perm, root0 + NEGO_N * 128, NEGO_N); // neg
    build_adj<<<(EEGO_N + 255) / 256, blk, 0, stream>>>(ego_src, ego_dst, adjbits);
    gemm_n128_wmma<<<(NEGO_N + 127) / 128, blk, 0, stream>>>(xego, BtWfc, bfc, fcx, NEGO_N, 128, 0);
    disc_scan<<<2, 128, NEGO_N * 128 * sizeof(float), stream>>>(fcx, Wfc, bfc, adjbits, snapm, snapri);
    build_hcat<<<(2 * EEGO_N * 128 + 255) / 256, blk, 0, stream>>>(
        root0, snapm, snapri, xego, ego_src, ego_dst, hcat);
    gemm_n128_wmma<<<(2 * EEGO_N + 127) / 128, blk, 0, stream>>>(hcat, BtWlin, blin, Hrelu, 2 * EEGO_N, 384, 1);
    logits_reduce<<<(2 * EEGO_N + 7) / 8, blk, 0, stream>>>(Hrelu, Wus, bus, sums);
    finalize<<<1, 1, 0, stream>>>(sums, (float*)d_out);
}